// NarrativeKGModel_58926951301723
// MI455X (gfx1250) — compile-verified
//
#include <hip/hip_runtime.h>
#include <hip/hip_bf16.h>

typedef __attribute__((ext_vector_type(2))) float v2f;
typedef __attribute__((ext_vector_type(8))) float v8f;
typedef __attribute__((ext_vector_type(4))) unsigned int v4u;
typedef __attribute__((ext_vector_type(8))) int v8i;
typedef __attribute__((ext_vector_type(4))) int v4i;

#define RGCN_R 8
#define RGCN_F 128
#define RGCN_H 128

// ---------------------------------------------------------------------------
// Scatter phase: one wave32 per edge. Gather x[src] (128 f32 = 4 per lane via
// float4) and atomically accumulate into sums[dst*R+etype]. Counts only on
// the first layer (graph topology identical for layer 2).
// ---------------------------------------------------------------------------
__global__ __launch_bounds__(256) void rgcn_scatter(
    const float* __restrict__ x,
    const int* __restrict__ src,
    const int* __restrict__ dst,
    const int* __restrict__ etype,
    float* __restrict__ sums,
    float* __restrict__ cnt,
    int nedges, int add_cnt)
{
    int wave = (int)((blockIdx.x * blockDim.x + threadIdx.x) >> 5);
    int lane = (int)(threadIdx.x & 31);
    if (wave >= nedges) return;

    int s   = src[wave];
    int seg = dst[wave] * RGCN_R + etype[wave];

    const float4* xr = (const float4*)(x + (size_t)s * RGCN_F);
    float4 v = xr[lane];

    float* o = sums + (size_t)seg * RGCN_F + lane * 4;
    atomicAdd(o + 0, v.x);
    atomicAdd(o + 1, v.y);
    atomicAdd(o + 2, v.z);
    atomicAdd(o + 3, v.w);
    if (add_cnt && lane == 0) atomicAdd(cnt + seg, 1.0f);
}

// ---------------------------------------------------------------------------
// Tensor Data Mover: DMA one 128x128 f32 weight matrix (64 KB, contiguous,
// row stride 128) from global memory into LDS at byte offset lds_off.
// D# layout per CDNA5 ISA ch.8 (group0: count/lds/global/type, group1: dims).
// 6-arg builtin form (amdgpu-toolchain clang-23).
// ---------------------------------------------------------------------------
__device__ __forceinline__ void tdm_load_weights(unsigned lds_off, const float* gsrc)
{
#if __has_builtin(__builtin_amdgcn_tensor_load_to_lds)
    unsigned long long ga = (unsigned long long)(uintptr_t)gsrc;
    v4u g0;
    g0[0] = 1u;                                          // count=1 (valid user D#)
    g0[1] = lds_off;                                     // lds_addr (bytes)
    g0[2] = (unsigned)(ga & 0xFFFFFFFFu);                // global_addr[31:0]
    g0[3] = (unsigned)((ga >> 32) & 0x01FFFFFFu)         // global_addr[56:32]
          | (2u << 30);                                  // type=2 ("image")
    v8i g1;
    g1[0] = (int)(2u << 16);        // workgroup_mask=0, data_size=2 (4 bytes)
    g1[1] = (int)(128u << 16);      // tensor_dim0 = 128 (low 16 in [31:16])
    g1[2] = (int)(128u << 16);      // tensor_dim0 hi=0 | tensor_dim1 = 128
    g1[3] = (int)(128u << 16);      // tensor_dim1 hi=0 | tile_dim0 = 128
    g1[4] = (int)(128u);            // tile_dim1 = 128, tile_dim2 = 0
    g1[5] = (int)(128u);            // tensor_dim0_stride = 128 (low 32)
    g1[6] = 0;                      // stride0 hi | tensor_dim1_stride lo
    g1[7] = 0;
    v4i z4 = {0, 0, 0, 0};
    v8i z8 = {0, 0, 0, 0, 0, 0, 0, 0};
    __builtin_amdgcn_tensor_load_to_lds(g0, g1, z4, z4, z8, 0);
#else
    (void)lds_off; (void)gsrc;
#endif
}

__device__ __forceinline__ void tdm_wait()
{
#if __has_builtin(__builtin_amdgcn_s_wait_tensorcnt)
    __builtin_amdgcn_s_wait_tensorcnt(0);
#endif
}

// ---------------------------------------------------------------------------
// GEMM phase: out[n,H] = sum_r mean_r @ relW[r] + xin @ rootW + bias (+ReLU).
// One wave32 computes a 16x128 output strip with V_WMMA_F32_16X16X4_F32.
// Weights are double-buffered in LDS via TDM (issued by wave 0, overlapped
// with compute); B fragments come from LDS as conflict-free ds_load_b32.
//   A frag (16x4, 2 VGPR): row m=lane&15, K=(lane>>4)*2+{0,1}
//   C/D (16x16, 8 VGPR):   VGPR v -> row v+8*(lane>>4), col=lane&15
// ---------------------------------------------------------------------------
__global__ __launch_bounds__(256) void rgcn_gemm(
    const float* __restrict__ sums,   // [N*R, F]
    const float* __restrict__ cnt,    // [N*R]
    const float* __restrict__ xin,    // [N, F]
    const float* __restrict__ relW,   // [R, F, H]
    const float* __restrict__ rootW,  // [F, H]
    const float* __restrict__ bias,   // [H]
    float* __restrict__ out,          // [N, H]
    int n, int do_relu)
{
    __shared__ float Wlds[2][RGCN_F * RGCN_H];           // 2 x 64 KB

    int wave = (int)((blockIdx.x * blockDim.x + threadIdx.x) >> 5);
    int lane = (int)(threadIdx.x & 31);
    int row0 = wave * 16;                                 // wave-uniform

    const int m  = lane & 15;
    const int kg = (lane >> 4) << 1;
    const bool issuer = (threadIdx.x < 32);               // wave 0 runs the TDM

    // weight source for "relation" r (r==R is the root transform)
    auto wsrc = [&](int r) -> const float* {
        return (r < RGCN_R) ? (relW + (size_t)r * RGCN_F * RGCN_H) : rootW;
    };
    // low 32 bits of a flat LDS pointer == LDS byte offset (ISA 10.2 aperture map)
    auto ldsoff = [&](int buf) -> unsigned {
        return (unsigned)(uintptr_t)(&Wlds[buf][0]);
    };

#if __has_builtin(__builtin_amdgcn_tensor_load_to_lds)
    const bool use_tdm = true;
#else
    const bool use_tdm = false;
#endif

    // Prologue: stage W(0) into buffer 0.
    if (use_tdm) {
        if (issuer) { tdm_load_weights(ldsoff(0), wsrc(0)); tdm_wait(); }
    } else {
        for (int i = threadIdx.x; i < RGCN_F * RGCN_H; i += 256)
            Wlds[0][i] = wsrc(0)[i];
    }
    __syncthreads();

    v8f acc[8];
#pragma unroll
    for (int t = 0; t < 8; ++t) acc[t] = (v8f)0.0f;

    for (int r = 0; r <= RGCN_R; ++r) {
        const int buf = r & 1;

        // Kick off the DMA for the next weight matrix into the other buffer.
        if (r < RGCN_R) {
            if (use_tdm) {
                if (issuer) tdm_load_weights(ldsoff(buf ^ 1), wsrc(r + 1));
            } else {
                for (int i = threadIdx.x; i < RGCN_F * RGCN_H; i += 256)
                    Wlds[buf ^ 1][i] = wsrc(r + 1)[i];
            }
        }

        const float* Arow;
        float scale;
        if (r < RGCN_R) {
            int seg = (row0 + m) * RGCN_R + r;
            Arow  = sums + (size_t)seg * RGCN_F;
            scale = (row0 < n) ? 1.0f / fmaxf(cnt[seg], 1.0f) : 0.0f;
        } else {
            Arow  = xin + (size_t)(row0 + m) * RGCN_F;
            scale = 1.0f;
        }
        if (row0 >= n) Arow = sums;                       // tail waves: keep EXEC
                                                          // full, read dummy row 0
        const float* Wl = &Wlds[buf][0];

#pragma unroll 4
        for (int k0 = 0; k0 < RGCN_F; k0 += 4) {
            v2f a;
            a.x = Arow[k0 + kg]     * scale;
            a.y = Arow[k0 + kg + 1] * scale;
#pragma unroll
            for (int t = 0; t < 8; ++t) {
                v2f b;
                b.x = Wl[(k0 + kg)     * RGCN_H + t * 16 + m];
                b.y = Wl[(k0 + kg + 1) * RGCN_H + t * 16 + m];
                acc[t] = __builtin_amdgcn_wmma_f32_16x16x4_f32(
                    false, a, false, b, (short)0, acc[t], false, false);
            }
        }

        if (r < RGCN_R) {
            if (use_tdm && issuer) tdm_wait();            // drain next-buffer DMA
            __syncthreads();
        }
    }

    if (row0 >= n) return;

    const int rbase = row0 + ((lane >> 4) << 3);
#pragma unroll
    for (int t = 0; t < 8; ++t) {
        int col = t * 16 + m;
        float bv = bias[col];
#pragma unroll
        for (int v = 0; v < 8; ++v) {
            float val = acc[t][v] + bv;
            if (do_relu) val = fmaxf(val, 0.0f);
            out[(size_t)(rbase + v) * RGCN_H + col] = val;
        }
    }
}

__global__ void rgcn_copy(const float* __restrict__ srcp, float* __restrict__ dstp, int n)
{
    int i = (int)(blockIdx.x * blockDim.x + threadIdx.x);
    if (i < n) dstp[i] = srcp[i];
}

extern "C" void kernel_launch(void* const* d_in, const int* in_sizes, int n_in,
                              void* d_out, int out_size, void* d_ws, size_t ws_size,
                              hipStream_t stream) {
    const float* x       = (const float*)d_in[0];
    const int*   eidx    = (const int*)d_in[1];
    const int*   etype   = (const int*)d_in[2];
    const float* rel_w0  = (const float*)d_in[3];
    const float* root_w0 = (const float*)d_in[4];
    const float* bias0   = (const float*)d_in[5];
    const float* rel_w1  = (const float*)d_in[6];
    const float* root_w1 = (const float*)d_in[7];
    const float* bias1   = (const float*)d_in[8];
    const float* rel_emb = (const float*)d_in[9];

    const int N = in_sizes[0] / RGCN_F;     // 50000
    const int E = in_sizes[2];              // 800000
    const int* src = eidx;
    const int* dst = eidx + E;

    float* out = (float*)d_out;

    float* sums = (float*)d_ws;                           // N*R*F f32
    float* cnt  = sums + (size_t)N * RGCN_R * RGCN_F;     // N*R f32
    float* h1   = cnt  + (size_t)N * RGCN_R;              // N*H f32

    const size_t sums_bytes = (size_t)N * RGCN_R * RGCN_F * sizeof(float);
    const size_t cnt_bytes  = (size_t)N * RGCN_R * sizeof(float);

    dim3 blk(256);
    dim3 grd_scatter((unsigned)(((size_t)E * 32 + 255) / 256));
    int nwaves = (N + 15) / 16;                           // 3125
    dim3 grd_gemm((unsigned)((nwaves * 32 + 255) / 256));

    // ---- Layer 1 ----
    (void)hipMemsetAsync(sums, 0, sums_bytes + cnt_bytes, stream);
    rgcn_scatter<<<grd_scatter, blk, 0, stream>>>(x, src, dst, etype, sums, cnt, E, 1);
    rgcn_gemm<<<grd_gemm, blk, 0, stream>>>(sums, cnt, x, rel_w0, root_w0, bias0, h1,
                                            N, /*relu=*/1);

    // ---- Layer 2 (counts unchanged: same graph) ----
    (void)hipMemsetAsync(sums, 0, sums_bytes, stream);
    rgcn_scatter<<<grd_scatter, blk, 0, stream>>>(h1, src, dst, etype, sums, cnt, E, 0);
    rgcn_gemm<<<grd_gemm, blk, 0, stream>>>(sums, cnt, h1, rel_w1, root_w1, bias1, out,
                                            N, /*relu=*/0);

    // ---- rel_emb passthrough ----
    rgcn_copy<<<(RGCN_R * RGCN_H + 255) / 256, blk, 0, stream>>>(
        rel_emb, out + (size_t)N * RGCN_H, RGCN_R * RGCN_H);
}